// MaskedDotProductAttention_32435593020205
// MI455X (gfx1250) — compile-verified
//
#include <hip/hip_runtime.h>
#include <math.h>

typedef __attribute__((ext_vector_type(2))) float v2f;
typedef __attribute__((ext_vector_type(8))) float v8f;

#define B_BATCH  16
#define S_LEN    2048
#define D_DIM    256
#define QSTR     260    // Q tile row stride (floats), +4 pad kills LDS bank conflicts
#define PSTR     2052   // score strip row stride (floats), +4 pad
#define NTHREADS 128
#define NWAVES   4

// D = A(16x4) * B(4x16) + C, fp32 WMMA, accumulated over K.
// Per-lane operand layout (ISA 7.12.2):
//   A: lane L -> row M = L&15, vgpr j -> K = 2*(L>>4) + j
//   B: lane L -> col N = L&15, vgpr j -> K = 2*(L>>4) + j
//   C/D: lane L, vgpr v -> row M = v + 8*(L>>4), col N = L&15
__global__ __launch_bounds__(NTHREADS)
void attn_fused_f32wmma(const float* __restrict__ Q,
                        const float* __restrict__ Kp,
                        const float* __restrict__ Vp,
                        const int*   __restrict__ vlens,
                        float* __restrict__ attn_out,   // [B,S,S]
                        float* __restrict__ ctx_out)    // [B,S,D]
{
    extern __shared__ float smem[];
    float* qtile  = smem;                    // 16 * QSTR
    float* pstrip = smem + 16 * QSTR;        // 16 * PSTR  (score strip / prob strip)
    float* redbuf = pstrip + 16 * PSTR;      // [0..15]=row max, [16..31]=1/rowsum

    const int tid  = threadIdx.x;
    const int lane = tid & 31;
    const int wave = tid >> 5;

    const int batch = blockIdx.x >> 7;           // 128 query tiles per batch
    const int qb    = (blockIdx.x & 127) * 16;   // first query row of this tile

    const float* Qb = Q  + (size_t)batch * S_LEN * D_DIM;
    const float* Kb = Kp + (size_t)batch * S_LEN * D_DIM;
    const float* Vb = Vp + (size_t)batch * S_LEN * D_DIM;
    const int vlen = vlens[batch];

    // ---- stage Q tile (16 x 256) into LDS, coalesced ----
    for (int idx = tid; idx < 16 * D_DIM; idx += NTHREADS) {
        int r = idx >> 8, c = idx & (D_DIM - 1);
        qtile[r * QSTR + c] = Qb[(size_t)(qb + r) * D_DIM + c];
    }
    __syncthreads();

    const int mrow  = lane & 15;         // A row / B col for this lane
    const int khalf = (lane >> 4) * 2;   // K sub-offset per lane half
    const int crow0 = (lane >> 4) * 8;   // first C/D row for this lane half

    // ================= Phase 1: scores = Q K^T / sqrt(D), masked ================
    for (int nt = wave; nt < S_LEN / 16; nt += NWAVES) {
        v8f acc = {0.f, 0.f, 0.f, 0.f, 0.f, 0.f, 0.f, 0.f};
        const float* arow = qtile + mrow * QSTR + khalf;                       // LDS
        const float* brow = Kb + (size_t)(nt * 16 + mrow) * D_DIM + khalf;     // global
        for (int k0 = 0; k0 < D_DIM; k0 += 4) {
            v2f a = *(const v2f*)(arow + k0);
            v2f b = *(const v2f*)(brow + k0);
            acc = __builtin_amdgcn_wmma_f32_16x16x4_f32(
                false, a, false, b, (short)0, acc, false, false);
        }
        const int col   = nt * 16 + mrow;        // key index of this lane's column
        const bool keep = (col < vlen);
        #pragma unroll
        for (int v = 0; v < 8; ++v) {
            float s = keep ? acc[v] * 0.0625f : -1000000.0f;  // 1/sqrt(256)
            pstrip[(crow0 + v) * PSTR + col] = s;             // coalesced in LDS
        }
    }
    __syncthreads();

    // ================= Phase 2a: per-row max and sum (8 threads / row) =========
    {
        const int row = tid >> 3;
        const int g   = tid & 7;
        const float* prow = pstrip + row * PSTR + g * 256;
        float mx = -3.0e38f;
        for (int c = 0; c < 256; ++c) mx = fmaxf(mx, prow[c]);
        mx = fmaxf(mx, __shfl_xor(mx, 4, 8));
        mx = fmaxf(mx, __shfl_xor(mx, 2, 8));
        mx = fmaxf(mx, __shfl_xor(mx, 1, 8));
        float sum = 0.f;
        for (int c = 0; c < 256; ++c) sum += __expf(prow[c] - mx);
        sum += __shfl_xor(sum, 4, 8);
        sum += __shfl_xor(sum, 2, 8);
        sum += __shfl_xor(sum, 1, 8);
        if (g == 0) { redbuf[row] = mx; redbuf[16 + row] = 1.0f / sum; }
    }
    __syncthreads();

    // ===== Phase 2b: normalize; write weights to global (coalesced) + LDS ======
    float* attnb = attn_out + ((size_t)batch * S_LEN + qb) * S_LEN;
    for (int idx = tid; idx < 16 * S_LEN; idx += NTHREADS) {
        int row = idx >> 11;
        int col = idx & (S_LEN - 1);
        float w = __expf(pstrip[row * PSTR + col] - redbuf[row]) * redbuf[16 + row];
        pstrip[row * PSTR + col] = w;
        attnb[(size_t)row * S_LEN + col] = w;
    }
    __syncthreads();

    // ================= Phase 3: contexts = P @ V ===============================
    const int d0   = wave * 64;      // each wave owns 64 of the 256 output dims
    const int ncol = lane & 15;
    v8f c0 = {0.f,0.f,0.f,0.f,0.f,0.f,0.f,0.f};
    v8f c1 = c0, c2 = c0, c3 = c0;
    for (int k0 = 0; k0 < S_LEN; k0 += 4) {
        v2f a = *(const v2f*)(pstrip + mrow * PSTR + k0 + khalf);
        const float* vrow = Vb + (size_t)(k0 + khalf) * D_DIM + d0 + ncol;
        v2f b0, b1, b2, b3;
        b0.x = vrow[0];  b0.y = vrow[0  + D_DIM];
        b1.x = vrow[16]; b1.y = vrow[16 + D_DIM];
        b2.x = vrow[32]; b2.y = vrow[32 + D_DIM];
        b3.x = vrow[48]; b3.y = vrow[48 + D_DIM];
        c0 = __builtin_amdgcn_wmma_f32_16x16x4_f32(false, a, false, b0, (short)0, c0, false, false);
        c1 = __builtin_amdgcn_wmma_f32_16x16x4_f32(false, a, false, b1, (short)0, c1, false, false);
        c2 = __builtin_amdgcn_wmma_f32_16x16x4_f32(false, a, false, b2, (short)0, c2, false, false);
        c3 = __builtin_amdgcn_wmma_f32_16x16x4_f32(false, a, false, b3, (short)0, c3, false, false);
    }
    float* ctxb = ctx_out + ((size_t)batch * S_LEN + qb) * D_DIM + d0;
    #pragma unroll
    for (int v = 0; v < 8; ++v) {
        size_t ro = (size_t)(crow0 + v) * D_DIM;
        ctxb[ro + 0  + ncol] = c0[v];
        ctxb[ro + 16 + ncol] = c1[v];
        ctxb[ro + 32 + ncol] = c2[v];
        ctxb[ro + 48 + ncol] = c3[v];
    }
}

extern "C" void kernel_launch(void* const* d_in, const int* in_sizes, int n_in,
                              void* d_out, int out_size, void* d_ws, size_t ws_size,
                              hipStream_t stream)
{
    const float* Q  = (const float*)d_in[0];
    const float* K  = (const float*)d_in[1];
    const float* V  = (const float*)d_in[2];
    const int*   vl = (const int*)d_in[3];

    float* attn = (float*)d_out;                                   // [B,S,S]
    float* ctx  = attn + (size_t)B_BATCH * S_LEN * S_LEN;          // [B,S,D]

    // ~148 KB dynamic LDS: legal on CDNA5 (320 KB per workgroup), 2 WGs/WGP.
    size_t smem = (size_t)(16 * QSTR + 16 * PSTR + 32) * sizeof(float);
    dim3 grid(B_BATCH * (S_LEN / 16));   // 2048 workgroups
    attn_fused_f32wmma<<<grid, dim3(NTHREADS), smem, stream>>>(Q, K, V, vl, attn, ctx);
}